// ActiveInference_67559835566640
// MI455X (gfx1250) — compile-verified
//
#include <hip/hip_runtime.h>
#include <hip/hip_bf16.h>
#include <math.h>

#define S_DIM 64
#define K_DIM 16
#define O_DIM 32
#define H_DIM 30
#define T_DIM 200
#define N_DIM 512
#define P_DIM 69761           // S*O*2 + K*S*S + S + S + 1
#define EPS_C 1e-6f
#define LOG2PI 1.8378770664093453f

typedef __attribute__((ext_vector_type(2))) float v2f;
typedef __attribute__((ext_vector_type(4))) float v4f;
typedef __attribute__((ext_vector_type(8))) float v8f;

// ---------------------------------------------------------------------------
// Kernel 1: parameter transform. One block per batch element n.
// Produces: Bws [N,K,S,S] (softmax), rws [N,K,S] (-kl-eh),
// Bmat [N,O,S,2] = interleaved (w=0.5/var, g=mu/var) pairs for the WMMA GEMM,
// cst [N,S], pdf [N,H], binit [N,S] (softmax of D logits).
// ---------------------------------------------------------------------------
__global__ __launch_bounds__(256) void transform_kernel(
    const float* __restrict__ theta,
    float* __restrict__ Bws, float* __restrict__ rws,
    float* __restrict__ Bmat, float* __restrict__ cst,
    float* __restrict__ pdfws, float* __restrict__ binit)
{
    const int n = blockIdx.x;
    const int tid = threadIdx.x;
    __shared__ float ent_s[S_DIM], logCp_s[S_DIM], cbuf[S_DIM], dbuf[S_DIM];
    __shared__ float redC[2], redD[2];

    const float* th = theta + (size_t)n * P_DIM;

    if (tid >= 64 && tid < 128)  cbuf[tid - 64]  = th[69632 + (tid - 64)];
    if (tid >= 128 && tid < 192) dbuf[tid - 128] = th[69696 + (tid - 128)];
    if (tid == 192) {
        float ttau = th[P_DIM - 1];
        float tau = 1.0f / (1.0f + expf(-ttau)) * (2.0f * (float)H_DIM) + 1.0f;
        float lt = logf(tau);
        float s = 0.0f;
        float lp[H_DIM];
        for (int h = 0; h < H_DIM; ++h) {
            lp[h] = (float)h * lt - tau - lgammaf((float)h + 1.0f);
            s += expf(lp[h]);
        }
        float inv = 1.0f / s;
        for (int h = 0; h < H_DIM; ++h) pdfws[n * H_DIM + h] = expf(lp[h]) * inv;
    }
    if (tid < 64) {
        const int i = tid;
        float entsum = 0.0f, cstsum = 0.0f;
        for (int oo = 0; oo < O_DIM; ++oo) {
            float mu = th[i * O_DIM + oo];
            float lv = th[S_DIM * O_DIM + i * O_DIM + oo];
            float sp = (lv > 20.0f) ? lv : log1pf(expf(lv));
            float sd = sp + EPS_C;                  // A_std
            float inv_var = 1.0f / (sd * sd);
            float w = 0.5f * inv_var;
            float g = mu * inv_var;
            // pair-interleaved: Bmat[n][oo][i][2] = {w, g}
            Bmat[(size_t)n * 4096 + (size_t)oo * 128 + i * 2 + 0] = w;
            Bmat[(size_t)n * 4096 + (size_t)oo * 128 + i * 2 + 1] = g;
            float ls = logf(sd);
            entsum += 0.5f + 0.5f * LOG2PI + ls;
            cstsum += -mu * mu * w - ls;
        }
        ent_s[i] = entsum;
        cst[n * S_DIM + i] = cstsum - 0.5f * (float)O_DIM * LOG2PI;
    }
    __syncthreads();
    if (tid == 0) {
        float mC = -1e30f, mD = -1e30f;
        for (int j = 0; j < S_DIM; ++j) { mC = fmaxf(mC, cbuf[j]); mD = fmaxf(mD, dbuf[j]); }
        float sC = 0.0f, sD = 0.0f;
        for (int j = 0; j < S_DIM; ++j) { sC += expf(cbuf[j] - mC); sD += expf(dbuf[j] - mD); }
        redC[0] = mC; redC[1] = sC; redD[0] = mD; redD[1] = sD;
    }
    __syncthreads();
    if (tid < 64) {
        float Cj = expf(cbuf[tid] - redC[0]) / redC[1];
        logCp_s[tid] = logf(Cj + EPS_C);
        binit[n * S_DIM + tid] = expf(dbuf[tid] - redD[0]) / redD[1];
    }
    __syncthreads();
    // B softmax rows + reward r = -kl - eh ; rows indexed row = k*S + i
    for (int row = tid; row < K_DIM * S_DIM; row += 256) {
        const float* lg = th + 4096 + (size_t)row * S_DIM;
        float mx = -1e30f;
        for (int j = 0; j < S_DIM; ++j) mx = fmaxf(mx, lg[j]);
        float sum = 0.0f;
        for (int j = 0; j < S_DIM; ++j) sum += expf(lg[j] - mx);
        float inv = 1.0f / sum;
        float kl = 0.0f, eh = 0.0f;
        float* Brow = Bws + (size_t)n * 65536 + (size_t)row * S_DIM;
        for (int j = 0; j < S_DIM; ++j) {
            float Bj = expf(lg[j] - mx) * inv;
            Brow[j] = Bj;
            kl += Bj * (logf(Bj + EPS_C) - logCp_s[j]);
            eh += Bj * ent_s[j];
        }
        rws[(size_t)n * 1024 + row] = -kl - eh;
    }
}

// ---------------------------------------------------------------------------
// Kernel 2: Gaussian log-likelihood GEMM via V_WMMA_F32_16X16X4_F32.
// llh[t,n,i] = sum_m (-o_t[m]^2 * w[i,m] + o_t[m] * g[i,m]) + cst[n,i].
// One wave per 16(t) x 16(i) tile; K-reduction permuted so that at step ko
// lane-half hi consumes obs m = hi*16+ko:
//   A fragment (f32 16x4): lanes 0-15 hold K slots 0,1; lanes 16-31 slots 2,3
//   -> a = {-o_m^2, o_m}, all 16 obs per lane preloaded contiguously (4xb128).
//   B fragment: pair (w,g) for (m, col) is one aligned b64 load.
// t-index clamped instead of guarded: bogus A rows only feed D rows that the
// guarded store drops, so EXEC stays all-ones with no branches in the loop.
// ---------------------------------------------------------------------------
__global__ __launch_bounds__(256) void llh_wmma_kernel(
    const float* __restrict__ o,      // [T,N,O]
    const float* __restrict__ Bmat,   // [N,O,S,2]
    const float* __restrict__ cst,    // [N,S]
    float* __restrict__ llh)          // [T,N,S]
{
    const int wid = blockIdx.x * (blockDim.x >> 5) + (threadIdx.x >> 5);
    const int lane = threadIdx.x & 31;
    const int n = wid / 52;
    const int rem = wid % 52;
    const int t_tile = rem >> 2;      // 0..12 (13 tiles cover T=200)
    const int i_tile = rem & 3;       // 0..3
    const int row = lane & 15;
    const int hi = lane >> 4;

    const int t0 = t_tile * 16 + row;
    const int t0c = (t0 < T_DIM) ? t0 : (T_DIM - 1);   // clamp, no exec games
    const int col = i_tile * 16 + row;

    // Preload this lane's 16 contiguous obs values (m = hi*16 + 0..15).
    const float* orow = o + ((size_t)t0c * N_DIM + n) * O_DIM + hi * 16;
    v4f av[4];
    av[0] = *(const v4f*)(orow + 0);
    av[1] = *(const v4f*)(orow + 4);
    av[2] = *(const v4f*)(orow + 8);
    av[3] = *(const v4f*)(orow + 12);

    // B pair base for this lane: (m = hi*16, col), stride 128 floats per ko.
    const float* Bp = Bmat + (size_t)n * 4096 + (size_t)hi * 2048 + col * 2;

    v8f acc = {};
#pragma unroll
    for (int ko = 0; ko < 16; ++ko) {
        float ov = av[ko >> 2][ko & 3];
        v2f a; a.x = -ov * ov; a.y = ov;
        v2f b = *(const v2f*)(Bp + (size_t)ko * 128);
        acc = __builtin_amdgcn_wmma_f32_16x16x4_f32(
            false, a, false, b, (short)0, acc, false, false);
    }
    const int i = i_tile * 16 + row;
    const float c = cst[n * S_DIM + i];
#pragma unroll
    for (int vv = 0; vv < 8; ++vv) {
        const int m = vv + hi * 8;
        const int t = t_tile * 16 + m;
        if (t < T_DIM)
            llh[((size_t)t * N_DIM + n) * S_DIM + i] = acc[vv] + c;
    }
}

// ---------------------------------------------------------------------------
// Kernel 3: finite-horizon value iteration. One block per n.
// q[h,n,k,i]: q[0]=r, q[h] = r + B . logsumexp_k q[h-1]. LDS double buffer.
// ---------------------------------------------------------------------------
__global__ __launch_bounds__(256) void vi_kernel(
    const float* __restrict__ Bws, const float* __restrict__ rws,
    float* __restrict__ q)            // [H,N,K,S]
{
    const int n = blockIdx.x;
    const int tid = threadIdx.x;
    __shared__ float qbuf[2][K_DIM * S_DIM];
    __shared__ float v_s[S_DIM];
    const float* rn = rws + (size_t)n * 1024;

    for (int p = tid; p < K_DIM * S_DIM; p += 256) {
        float rv = rn[p];
        qbuf[0][p] = rv;
        q[((size_t)0 * N_DIM + n) * 1024 + p] = rv;
    }
    __syncthreads();
    int cur = 0;
    for (int h = 1; h < H_DIM; ++h) {
        if (tid < 64) {
            const int j = tid;
            float mx = -1e30f;
            for (int k = 0; k < K_DIM; ++k) mx = fmaxf(mx, qbuf[cur][k * S_DIM + j]);
            float s = 0.0f;
            for (int k = 0; k < K_DIM; ++k) s += expf(qbuf[cur][k * S_DIM + j] - mx);
            v_s[j] = mx + logf(s);
        }
        __syncthreads();
        const int nxt = cur ^ 1;
        for (int p = tid; p < K_DIM * S_DIM; p += 256) {
            const float* Brow = Bws + (size_t)n * 65536 + (size_t)p * S_DIM;
            float accv = 0.0f;
            for (int j = 0; j < S_DIM; ++j) accv += Brow[j] * v_s[j];
            float val = rn[p] + accv;
            qbuf[nxt][p] = val;
            q[((size_t)h * N_DIM + n) * 1024 + p] = val;
        }
        __syncthreads();
        cur = nxt;
    }
}

// ---------------------------------------------------------------------------
// Kernel 4: sequential belief scan over T. One block per n.
// ---------------------------------------------------------------------------
__global__ __launch_bounds__(256) void timeloop_kernel(
    const int* __restrict__ a,
    const float* __restrict__ Bws, const float* __restrict__ q,
    const float* __restrict__ llh, const float* __restrict__ pdfws,
    const float* __restrict__ binit,
    float* __restrict__ out_pi, float* __restrict__ out_b,
    float* __restrict__ out_lpo)
{
    const int n = blockIdx.x;
    const int tid = threadIdx.x;
    __shared__ float b_s[S_DIM], pdf_s[H_DIM], qb_s[H_DIM * K_DIM];
    __shared__ float hmax[H_DIM], hsum[H_DIM], pi_s[K_DIM];
    __shared__ float joint_s[S_DIM], red[2];

    if (tid < 64) b_s[tid] = binit[n * S_DIM + tid];
    if (tid < H_DIM) pdf_s[tid] = pdfws[n * H_DIM + tid];
    __syncthreads();

    for (int t = 0; t < T_DIM; ++t) {
        if (tid < 64) out_b[((size_t)t * N_DIM + n) * S_DIM + tid] = b_s[tid];
        if (tid < K_DIM) pi_s[tid] = 0.0f;
        // qb[h,k] = sum_i b[i] * q[h,n,k,i]
        for (int p = tid; p < H_DIM * K_DIM; p += 256) {
            const int h = p >> 4, k = p & 15;
            const float* qrow = q + ((size_t)h * N_DIM + n) * 1024 + k * S_DIM;
            float accv = 0.0f;
            for (int i = 0; i < S_DIM; ++i) accv += b_s[i] * qrow[i];
            qb_s[p] = accv;
        }
        __syncthreads();
        if (tid < H_DIM) {
            const int h = tid;
            float mx = -1e30f;
            for (int k = 0; k < K_DIM; ++k) mx = fmaxf(mx, qb_s[h * 16 + k]);
            float s = 0.0f;
            for (int k = 0; k < K_DIM; ++k) s += expf(qb_s[h * 16 + k] - mx);
            hmax[h] = mx; hsum[h] = s;
        }
        __syncthreads();
        for (int p = tid; p < H_DIM * K_DIM; p += 256) {
            const int h = p >> 4, k = p & 15;
            float val = pdf_s[h] * expf(qb_s[p] - hmax[h]) / hsum[h];
            atomicAdd(&pi_s[k], val);
        }
        __syncthreads();
        if (tid < K_DIM) out_pi[((size_t)t * N_DIM + n) * K_DIM + tid] = pi_s[tid];
        // belief update
        const int at = a[t * N_DIM + n];
        if (tid < 64) {
            const int j = tid;
            const float* Ba = Bws + (size_t)n * 65536 + (size_t)at * 4096;
            float s = 0.0f;
            for (int i = 0; i < S_DIM; ++i) s += Ba[i * S_DIM + j] * b_s[i];
            joint_s[j] = logf(s + EPS_C) + llh[((size_t)t * N_DIM + n) * S_DIM + j];
        }
        __syncthreads();
        if (tid == 0) {
            float mx = -1e30f;
            for (int j = 0; j < S_DIM; ++j) mx = fmaxf(mx, joint_s[j]);
            float s = 0.0f;
            for (int j = 0; j < S_DIM; ++j) s += expf(joint_s[j] - mx);
            red[0] = mx; red[1] = s;
            out_lpo[(size_t)t * N_DIM + n] = mx + logf(s);
        }
        __syncthreads();
        if (tid < 64) b_s[tid] = expf(joint_s[tid] - red[0]) / red[1];
        __syncthreads();
    }
}

extern "C" void kernel_launch(void* const* d_in, const int* in_sizes, int n_in,
                              void* d_out, int out_size, void* d_ws, size_t ws_size,
                              hipStream_t stream) {
    const float* o     = (const float*)d_in[0];   // [T,N,O]
    const int*   a     = (const int*)d_in[1];     // [T,N]
    const float* theta = (const float*)d_in[2];   // [N,P]

    float* ws    = (float*)d_ws;
    float* Bws   = ws;                         // N*K*S*S = 33,554,432
    float* rws   = Bws   + 33554432ull;        // N*K*S   =    524,288
    float* qbuf  = rws   + 524288ull;          // H*N*K*S = 15,728,640 (31,457,280 reserved)
    float* Bmat  = qbuf  + 31457280ull;        // N*O*S*2 =  2,097,152
    float* cst   = Bmat  + 2097152ull;         // N*S     =     32,768
    float* llh   = cst   + 32768ull;           // T*N*S   =  6,553,600
    float* pdf   = llh   + 6553600ull;         // N*H     =     15,360
    float* binit = pdf   + 15360ull;           // N*S     =     32,768

    float* out_pi  = (float*)d_out;                          // [T,N,K]
    float* out_b   = out_pi + (size_t)T_DIM * N_DIM * K_DIM; // [T,N,S]
    float* out_lpo = out_b  + (size_t)T_DIM * N_DIM * S_DIM; // [T,N]

    transform_kernel<<<N_DIM, 256, 0, stream>>>(theta, Bws, rws, Bmat, cst, pdf, binit);
    // N * 13 t-tiles * 4 i-tiles = 26624 waves / 8 waves per block
    llh_wmma_kernel<<<(N_DIM * 52) / 8, 256, 0, stream>>>(o, Bmat, cst, llh);
    vi_kernel<<<N_DIM, 256, 0, stream>>>(Bws, rws, qbuf);
    timeloop_kernel<<<N_DIM, 256, 0, stream>>>(a, Bws, qbuf, llh, pdf, binit,
                                               out_pi, out_b, out_lpo);
}